// GcnNet_8057358647517
// MI455X (gfx1250) — compile-verified
//
#include <hip/hip_runtime.h>

typedef __attribute__((ext_vector_type(2))) float v2f;
typedef __attribute__((ext_vector_type(8))) float v8f;

#define N_NODES 4096
#define N_GRAPHS 32
#define NTOT (N_NODES * N_GRAPHS)   // 131072
#define DEG 32
#define ETOT (NTOT * DEG)           // 4194304
#define COUT 20
#define LRELU(v) ((v) > 0.f ? (v) : 0.01f * (v))

// Async copy of one f32 from global memory into LDS (gfx1250 async-vmem path,
// tracked by ASYNCcnt). lds_off is the wave-relative LDS byte address.
__device__ __forceinline__ void async_copy_f32(unsigned int lds_off, const float* g) {
    asm volatile("global_load_async_to_lds_b32 %0, %1, off"
                 :: "v"(lds_off), "v"(g)
                 : "memory");
}
__device__ __forceinline__ void wait_async_all() {
    asm volatile("s_wait_asynccnt 0x0" ::: "memory");
}

__global__ void zero_f32(float* __restrict__ p, int n) {
    int i = blockIdx.x * blockDim.x + threadIdx.x;
    if (i < n) p[i] = 0.f;
}

// agg[dst, :] += x[src, :] * w   (one thread per edge; atomics resolve in L2 --
// the 10.5 MB accumulator fits the 192 MB L2 easily)
template <int C>
__global__ void scatter_edges(const float* __restrict__ x,
                              const int* __restrict__ ei,
                              const float* __restrict__ ea,
                              float* __restrict__ agg) {
    int e = blockIdx.x * blockDim.x + threadIdx.x;
    if (e >= ETOT) return;
    int src = ei[e];
    int dst = ei[ETOT + e];
    float w = ea[3 * e + 2];
    const float* xs = x + (long)src * C;
    float* ad = agg + (long)dst * C;
#pragma unroll
    for (int c = 0; c < C; ++c) {
        atomicAdd(ad + c, xs[c] * w);
    }
}

// H[n, 0..19] = leaky( AGG[n,:] @ Wrel + bias + X[n,:] @ Wroot )
// One wave per 16-row strip; two f32 WMMA accumulators cover cols 0..31
// (20 valid). Weights zero-padded to K x 32 in LDS via async global->LDS copy.
template <int K>
__global__ void gemm_conv_wmma(const float* __restrict__ AGG,
                               const float* __restrict__ Wrel,
                               const float* __restrict__ bias,
                               const float* __restrict__ X,
                               const float* __restrict__ Wroot,
                               float* __restrict__ H) {
    __shared__ float sWrel[K * 32];
    __shared__ float sWroot[K * 32];
    int t = threadIdx.x;   // blockDim.x == 128

    // Zero-fill pad columns, then async-fill the valid 20 columns.
    for (int i = t; i < K * 32; i += 128) { sWrel[i] = 0.f; sWroot[i] = 0.f; }
    __syncthreads();
    for (int i = t; i < K * COUT; i += 128) {
        int k = i / COUT, c = i - k * COUT;
        unsigned int off = (unsigned int)((k * 32 + c) * 4);
        async_copy_f32((unsigned int)(uintptr_t)sWrel + off,  Wrel + i);
        async_copy_f32((unsigned int)(uintptr_t)sWroot + off, Wroot + i);
    }
    wait_async_all();
    __syncthreads();

    int wave = t >> 5;
    int lane = t & 31;
    int half = lane >> 4;   // 0: lanes 0-15 (K=k,k+1), 1: lanes 16-31 (K=k+2,k+3)
    int l    = lane & 15;

    int tile = blockIdx.x * 4 + wave;   // 16-row strip index
    int m0 = tile * 16;
    int row = m0 + l;

    v8f acc0 = {};   // cols 0..15
    v8f acc1 = {};   // cols 16..31 (16..19 valid)

#pragma unroll
    for (int k = 0; k < K; k += 4) {
        int kb = k + 2 * half;
        v2f a = *(const v2f*)(AGG + (long)row * K + kb);
        v2f b0, b1;
        b0.x = sWrel[kb * 32 + l];
        b0.y = sWrel[(kb + 1) * 32 + l];
        b1.x = sWrel[kb * 32 + 16 + l];
        b1.y = sWrel[(kb + 1) * 32 + 16 + l];
        acc0 = __builtin_amdgcn_wmma_f32_16x16x4_f32(false, a, false, b0,
                                                     (short)0, acc0, false, false);
        acc1 = __builtin_amdgcn_wmma_f32_16x16x4_f32(false, a, false, b1,
                                                     (short)0, acc1, false, false);
    }
#pragma unroll
    for (int k = 0; k < K; k += 4) {
        int kb = k + 2 * half;
        v2f a = *(const v2f*)(X + (long)row * K + kb);
        v2f b0, b1;
        b0.x = sWroot[kb * 32 + l];
        b0.y = sWroot[(kb + 1) * 32 + l];
        b1.x = sWroot[kb * 32 + 16 + l];
        b1.y = sWroot[(kb + 1) * 32 + 16 + l];
        acc0 = __builtin_amdgcn_wmma_f32_16x16x4_f32(false, a, false, b0,
                                                     (short)0, acc0, false, false);
        acc1 = __builtin_amdgcn_wmma_f32_16x16x4_f32(false, a, false, b1,
                                                     (short)0, acc1, false, false);
    }

    // C/D layout: VGPR v -> row m0+v (+8 for upper half-lanes), col = n0 + l.
    float bc0 = bias[l];
    float bc1 = (l < 4) ? bias[16 + l] : 0.f;
#pragma unroll
    for (int v = 0; v < 8; ++v) {
        int r = m0 + v + half * 8;
        float v0 = acc0[v] + bc0;
        H[(long)r * COUT + l] = LRELU(v0);
        if (l < 4) {
            float v1 = acc1[v] + bc1;
            H[(long)r * COUT + 16 + l] = LRELU(v1);
        }
    }
}

// BatchNorm over axis (graphs, channels) per node position; in-place.
__global__ void bn_nodes(float* __restrict__ H,
                         const float* __restrict__ gamma,
                         const float* __restrict__ beta) {
    const int CNT = N_GRAPHS * COUT;   // 640
    int pos = blockIdx.x;
    int t = threadIdx.x;
    __shared__ float s1[128], s2[128];

    float sum = 0.f, sq = 0.f;
    for (int i = t; i < CNT; i += 128) {
        int g = i / COUT, c = i - (i / COUT) * COUT;
        float v = H[((long)(g * N_NODES + pos)) * COUT + c];
        sum += v; sq += v * v;
    }
    s1[t] = sum; s2[t] = sq;
    __syncthreads();
    for (int s = 64; s > 0; s >>= 1) {
        if (t < s) { s1[t] += s1[t + s]; s2[t] += s2[t + s]; }
        __syncthreads();
    }
    float mu  = s1[0] / (float)CNT;
    float var = s2[0] / (float)CNT - mu * mu;
    float inv = rsqrtf(var + 1e-5f);
    float ga = gamma[pos], be = beta[pos];
    for (int i = t; i < CNT; i += 128) {
        int g = i / COUT, c = i - (i / COUT) * COUT;
        long idx = ((long)(g * N_NODES + pos)) * COUT + c;
        H[idx] = (H[idx] - mu) * inv * ga + be;
    }
}

// Layer-3 conv (Cout=1) + leaky; writes h (flat N) straight into d_out.
__global__ void conv3(const float* __restrict__ AGG,
                      const float* __restrict__ Wrel,
                      const float* __restrict__ b3,
                      const float* __restrict__ X,
                      const float* __restrict__ Wroot,
                      float* __restrict__ hout) {
    int n = blockIdx.x * blockDim.x + threadIdx.x;
    if (n >= NTOT) return;
    float s = b3[0];
#pragma unroll
    for (int c = 0; c < COUT; ++c)
        s += AGG[(long)n * COUT + c] * Wrel[c] + X[(long)n * COUT + c] * Wroot[c];
    hout[n] = LRELU(s);
}

// Per-graph normalize (ddof=1, divide by var+eps per reference) + FC head.
__global__ void head(const float* __restrict__ h,
                     const float* __restrict__ fc1W, const float* __restrict__ fc1b,
                     const float* __restrict__ fc2W, const float* __restrict__ fc2b,
                     float* __restrict__ out) {
    int g = blockIdx.x;
    int t = threadIdx.x;
    const float* hr = h + (long)g * N_NODES;
    __shared__ float red[256];
    __shared__ float s_fc1[50];
    __shared__ float s_mu, s_inv;

    float sum = 0.f, sq = 0.f;
    for (int i = t; i < N_NODES; i += 256) { float v = hr[i]; sum += v; sq += v * v; }
    red[t] = sum; __syncthreads();
    for (int s = 128; s > 0; s >>= 1) { if (t < s) red[t] += red[t + s]; __syncthreads(); }
    float mu = red[0] / (float)N_NODES;
    __syncthreads();
    red[t] = sq; __syncthreads();
    for (int s = 128; s > 0; s >>= 1) { if (t < s) red[t] += red[t + s]; __syncthreads(); }
    if (t == 0) {
        float var = (red[0] - (float)N_NODES * mu * mu) / (float)(N_NODES - 1);
        s_mu = mu;
        s_inv = 1.f / (var + 1e-10f);
    }
    __syncthreads();
    float m = s_mu, inv = s_inv;

    for (int j = 0; j < 50; ++j) {
        float p = 0.f;
        for (int i = t; i < N_NODES; i += 256)
            p += (hr[i] - m) * inv * fc1W[(long)i * 50 + j];
        red[t] = p; __syncthreads();
        for (int s = 128; s > 0; s >>= 1) { if (t < s) red[t] += red[t + s]; __syncthreads(); }
        if (t == 0) s_fc1[j] = LRELU(red[0] + fc1b[j]);
        __syncthreads();
    }
    if (t < 2) {
        float o = fc2b[t];
#pragma unroll
        for (int j = 0; j < 50; ++j) o += s_fc1[j] * fc2W[j * 2 + t];
        out[g * 2 + t] = o;
    }
}

extern "C" void kernel_launch(void* const* d_in, const int* in_sizes, int n_in,
                              void* d_out, int out_size, void* d_ws, size_t ws_size,
                              hipStream_t stream) {
    const float* x      = (const float*)d_in[0];
    const int*   ei     = (const int*)d_in[1];
    const float* ea     = (const float*)d_in[2];
    const float* Wrel1  = (const float*)d_in[3];
    const float* b1     = (const float*)d_in[4];
    const float* Wroot1 = (const float*)d_in[5];
    const float* Wrel2  = (const float*)d_in[6];
    const float* b2     = (const float*)d_in[7];
    const float* Wroot2 = (const float*)d_in[8];
    const float* Wrel3  = (const float*)d_in[9];
    const float* b3     = (const float*)d_in[10];
    const float* Wroot3 = (const float*)d_in[11];
    const float* bn1g   = (const float*)d_in[12];
    const float* bn1b   = (const float*)d_in[13];
    const float* bn2g   = (const float*)d_in[14];
    const float* bn2b   = (const float*)d_in[15];
    const float* fc1W   = (const float*)d_in[16];
    const float* fc1b   = (const float*)d_in[17];
    const float* fc2W   = (const float*)d_in[18];
    const float* fc2b   = (const float*)d_in[19];

    float* out  = (float*)d_out;
    float* hout = out + N_GRAPHS * 2;        // h occupies d_out[64 : 64+131072]

    float* ws   = (float*)d_ws;
    const size_t NB = (size_t)NTOT * COUT;   // 2.62M floats per buffer
    float* aggA = ws;
    float* H1   = ws + NB;
    float* aggB = ws + 2 * NB;
    float* H2   = ws + 3 * NB;

    const int gemm_blocks = (NTOT / 16) / 4;   // 8192 row strips, 4 waves/block

    // ---- layer 1 (K=16) ----
    zero_f32<<<(NTOT * 16 + 255) / 256, 256, 0, stream>>>(aggA, NTOT * 16);
    scatter_edges<16><<<(ETOT + 255) / 256, 256, 0, stream>>>(x, ei, ea, aggA);
    gemm_conv_wmma<16><<<gemm_blocks, 128, 0, stream>>>(aggA, Wrel1, b1, x, Wroot1, H1);
    bn_nodes<<<N_NODES, 128, 0, stream>>>(H1, bn1g, bn1b);

    // ---- layer 2 (K=20) ----
    zero_f32<<<(NTOT * 20 + 255) / 256, 256, 0, stream>>>(aggB, NTOT * 20);
    scatter_edges<20><<<(ETOT + 255) / 256, 256, 0, stream>>>(H1, ei, ea, aggB);
    gemm_conv_wmma<20><<<gemm_blocks, 128, 0, stream>>>(aggB, Wrel2, b2, H1, Wroot2, H2);
    bn_nodes<<<N_NODES, 128, 0, stream>>>(H2, bn2g, bn2b);

    // ---- layer 3 (Cout=1) ----
    zero_f32<<<(NTOT * 20 + 255) / 256, 256, 0, stream>>>(aggA, NTOT * 20);
    scatter_edges<20><<<(ETOT + 255) / 256, 256, 0, stream>>>(H2, ei, ea, aggA);
    conv3<<<(NTOT + 255) / 256, 256, 0, stream>>>(aggA, Wrel3, b3, H2, Wroot3, hout);

    // ---- head ----
    head<<<N_GRAPHS, 256, 0, stream>>>(hout, fc1W, fc1b, fc2W, fc2b, out);
}